// NodeFeatures_32633161515440
// MI455X (gfx1250) — compile-verified
//
#include <hip/hip_runtime.h>

#define EPSV 1e-20f

typedef __attribute__((ext_vector_type(2))) float v2f;
typedef __attribute__((ext_vector_type(8))) float v8f;

// Problem shape fixed by the reference
constexpr int Bn = 8;
constexpr int Nn = 256;
constexpr int Hn = 128;
constexpr int Mn = Bn * Nn;          // 2048 rows

// ---------------------------------------------------------------------------
// Kernel 1: Ux = mask .* (x @ Uw^T + Ub), Vx = mask .* (x @ Vw^T + Vb)
// One wave32 per 16x16 output tile; V_WMMA_F32_16X16X4_F32 over K=128.
// A layout (16x4 f32): lane l -> M=l%16; VGPR v, group g=l/16 -> K=v+2g.
// B layout (4x16 f32): lane l -> N=l%16; VGPR v, group g    -> K=v+2g.
//   B[k][n] = W[n][k]  (W row-major (o,h), reduction over h) -> per-lane
//   float2 load from W[(otile*16 + l%16)*H + k + 2g], contiguous in K.
// C/D layout: VGPR r, group g -> M = r + 8g, N = l%16.
// ---------------------------------------------------------------------------
__global__ __launch_bounds__(128) void gemm_uv_wmma(
    const float* __restrict__ x,
    const float* __restrict__ Uw, const float* __restrict__ Ub,
    const float* __restrict__ Vw, const float* __restrict__ Vb,
    const float* __restrict__ mask,
    float* __restrict__ Ux, float* __restrict__ Vx)
{
    const int wave = threadIdx.x >> 5;
    const int lane = threadIdx.x & 31;
    const int gid  = blockIdx.x * 4 + wave;   // 0..2047 (1024 tiles x 2 matrices)
    const int sel  = gid >> 10;               // 0 = U path, 1 = V path
    const int tile = gid & 1023;
    const int mt   = tile >> 3;               // 0..127  (M/16)
    const int ot   = tile & 7;                // 0..7    (H/16)

    const float* W    = sel ? Vw : Uw;
    const float* bias = sel ? Vb : Ub;
    float*       dst  = sel ? Vx : Ux;

    const int l16 = lane & 15;
    const int g   = lane >> 4;                // 0 or 1

    const float* arow = x + (size_t)(mt * 16 + l16) * Hn + 2 * g;
    const float* brow = W + (size_t)(ot * 16 + l16) * Hn + 2 * g;

    v8f acc = {0.f, 0.f, 0.f, 0.f, 0.f, 0.f, 0.f, 0.f};
#pragma unroll 8
    for (int k = 0; k < Hn; k += 4) {
        v2f a = *(const v2f*)(arow + k);
        v2f b = *(const v2f*)(brow + k);
        // D = A*B + C ; emits v_wmma_f32_16x16x4_f32
        acc = __builtin_amdgcn_wmma_f32_16x16x4_f32(
                  /*neg_a=*/false, a, /*neg_b=*/false, b,
                  /*c_mod=*/(short)0, acc, /*reuse_a=*/false, /*reuse_b=*/false);
    }

    const int   o  = ot * 16 + l16;
    const float bo = bias[o];
#pragma unroll
    for (int r = 0; r < 8; ++r) {
        const int m = mt * 16 + r + 8 * g;
        dst[(size_t)m * Hn + o] = mask[m] * (acc[r] + bo);
    }
}

// ---------------------------------------------------------------------------
// Kernel 2: streaming aggregation over edge_gate (the 268 MB HBM stream).
// Block = one (b,i) destination row, 256 threads = 8 waves.
// Wave jg handles j = jg, jg+8, ... ; its 32 lanes load one contiguous 512B
// row of eg / Vx as float4. Private accumulate, LDS-reduce across waves.
// out = Ux + (m_i * sum_j eg*m_j*Vx) / (EPS + m_i * sum_j eg*m_j)
// ---------------------------------------------------------------------------
__global__ __launch_bounds__(256) void edge_agg(
    const float* __restrict__ eg, const float* __restrict__ mask,
    const float* __restrict__ Ux, const float* __restrict__ Vx,
    float* __restrict__ out)
{
    const int bi = blockIdx.x;                // b*N + i
    const int b  = bi >> 8;
    const int i  = bi & 255;
    const int t  = threadIdx.x;
    const int h4 = t & 31;                    // owns h = 4*h4 .. 4*h4+3
    const int jg = t >> 5;                    // wave id = j-group 0..7

    __shared__ float sm[Nn];
    sm[t] = mask[b * Nn + t];
    __syncthreads();

    const float* egrow = eg + (size_t)bi * Nn * Hn + 4 * h4;
    const float* vbase = Vx + (size_t)b  * Nn * Hn + 4 * h4;

    float s1x = 0.f, s1y = 0.f, s1z = 0.f, s1w = 0.f;   // sum eg*mj*Vx
    float s0x = 0.f, s0y = 0.f, s0z = 0.f, s0w = 0.f;   // sum eg*mj
    for (int j = jg; j < Nn; j += 8) {
        const float4 e = *(const float4*)(egrow + (size_t)j * Hn);
        const float4 v = *(const float4*)(vbase + (size_t)j * Hn);
        // keep the HBM stream deep: prefetch 8KB ahead (speculative, OOB-safe)
        __builtin_prefetch(egrow + (size_t)(j + 16) * Hn, 0, 0);
        const float mj = sm[j];
        const float ex = e.x * mj, ey = e.y * mj, ez = e.z * mj, ew = e.w * mj;
        s0x += ex; s0y += ey; s0z += ez; s0w += ew;
        s1x = fmaf(ex, v.x, s1x);
        s1y = fmaf(ey, v.y, s1y);
        s1z = fmaf(ez, v.z, s1z);
        s1w = fmaf(ew, v.w, s1w);
    }

    __shared__ float4 r1[8][32];
    __shared__ float4 r0[8][32];
    r1[jg][h4] = make_float4(s1x, s1y, s1z, s1w);
    r0[jg][h4] = make_float4(s0x, s0y, s0z, s0w);
    __syncthreads();

    if (t < 32) {
        float4 a1 = r1[0][t];
        float4 a0 = r0[0][t];
#pragma unroll
        for (int g = 1; g < 8; ++g) {
            const float4 q1 = r1[g][t];
            const float4 q0 = r0[g][t];
            a1.x += q1.x; a1.y += q1.y; a1.z += q1.z; a1.w += q1.w;
            a0.x += q0.x; a0.y += q0.y; a0.z += q0.z; a0.w += q0.w;
        }
        const float  mi = sm[i];
        const float4 u  = *(const float4*)(Ux + (size_t)bi * Hn + 4 * t);
        float4 o;
        o.x = u.x + (mi * a1.x) / (EPSV + mi * a0.x);
        o.y = u.y + (mi * a1.y) / (EPSV + mi * a0.y);
        o.z = u.z + (mi * a1.z) / (EPSV + mi * a0.z);
        o.w = u.w + (mi * a1.w) / (EPSV + mi * a0.w);
        *(float4*)(out + (size_t)bi * Hn + 4 * t) = o;
    }
}

extern "C" void kernel_launch(void* const* d_in, const int* in_sizes, int n_in,
                              void* d_out, int out_size, void* d_ws, size_t ws_size,
                              hipStream_t stream) {
    const float* x    = (const float*)d_in[0];   // (B,N,H)
    const float* eg   = (const float*)d_in[1];   // (B,N,N,H)
    const float* mask = (const float*)d_in[2];   // (B,N)
    const float* Uw   = (const float*)d_in[3];   // (H,H)
    const float* Ub   = (const float*)d_in[4];   // (H)
    const float* Vw   = (const float*)d_in[5];   // (H,H)
    const float* Vb   = (const float*)d_in[6];   // (H)
    float* out = (float*)d_out;                  // (B,N,H)

    float* Ux = (float*)d_ws;                    // 2048*128 f32
    float* Vx = Ux + (size_t)Mn * Hn;            // 2048*128 f32 (2 MB total)

    // 1024 tiles per matrix * 2 matrices = 2048 waves; 4 waves/block -> 512 blocks
    gemm_uv_wmma<<<512, 128, 0, stream>>>(x, Uw, Ub, Vw, Vb, mask, Ux, Vx);
    // one block per (b,i) row
    edge_agg<<<Mn, 256, 0, stream>>>(eg, mask, Ux, Vx, out);
}